// SinkhornKnopp_66932770340885
// MI455X (gfx1250) — compile-verified
//
#include <hip/hip_runtime.h>

// ---------------------------------------------------------------------------
// Sinkhorn-Knopp on E = exp(Q/0.05), Q: 16384 x 3072 fp32 (row-major, b-major)
// Rank-1 factorized: Qt[k,b] = E[b,k] * U[k] * V[b]; E recomputed per pass
// (Q is 192 MiB == L2 size -> L2-resident re-reads; ~550 KB workspace only).
// ---------------------------------------------------------------------------

#define B_DIM 16384
#define K_DIM 3072
#define EPS_INV 20.0f            // 1/0.05
#define EXP_OVF 88.72283f        // ln(FLT_MAX): exp(a) overflows above this
#define R_TGT (1.0f / 3072.0f)   // row marginal r
#define C_TGT (1.0f / 16384.0f)  // col marginal c
#define SLABS 32                 // b-slabs for WMMA colsum pass (512 rows each)

typedef __attribute__((ext_vector_type(2))) float v2f;
typedef __attribute__((ext_vector_type(8))) float v8f;

// exp(q/eps) with inf -> (max finite entry) replacement folded in
__device__ __forceinline__ float expClamped(float q, float eMax) {
    float a = q * EPS_INV;
    float e = __expf(a);
    return (a > EXP_OVF) ? eMax : e;
}

// ---------------------------------------------------------------------------
// Pass 0: per-block partials of {sum of finite exp, #inf entries, max finite arg}
// ---------------------------------------------------------------------------
__global__ void stats_kernel(const float4* __restrict__ Q4,
                             float* __restrict__ pSum,
                             float* __restrict__ pCnt,
                             float* __restrict__ pMax) {
    const int N4 = (B_DIM * K_DIM) / 4;  // 12582912
    float s = 0.f, cnt = 0.f, mx = -1e30f;
    int stride = gridDim.x * blockDim.x;
    for (int i = blockIdx.x * blockDim.x + threadIdx.x; i < N4; i += stride) {
        float4 q = Q4[i];
        float v[4] = {q.x, q.y, q.z, q.w};
#pragma unroll
        for (int j = 0; j < 4; ++j) {
            float a = v[j] * EPS_INV;
            if (a > EXP_OVF) {
                cnt += 1.0f;
            } else {
                s += __expf(a);
                mx = fmaxf(mx, a);
            }
        }
    }
    __shared__ float shS[256], shC[256], shM[256];
    int t = threadIdx.x;
    shS[t] = s; shC[t] = cnt; shM[t] = mx;
    __syncthreads();
    for (int o = 128; o > 0; o >>= 1) {
        if (t < o) {
            shS[t] += shS[t + o];
            shC[t] += shC[t + o];
            shM[t] = fmaxf(shM[t], shM[t + o]);
        }
        __syncthreads();
    }
    if (t == 0) {
        pSum[blockIdx.x] = shS[0];
        pCnt[blockIdx.x] = shC[0];
        pMax[blockIdx.x] = shM[0];
    }
}

// Deterministic single-block reduction of the 512 partials.
// scal[0] = S = total sum after inf->max replacement; scal[1] = maxFiniteArg
__global__ void reduce0_kernel(const float* __restrict__ pSum,
                               const float* __restrict__ pCnt,
                               const float* __restrict__ pMax,
                               float* __restrict__ scal) {
    __shared__ float shS[512], shC[512], shM[512];
    int t = threadIdx.x;
    shS[t] = pSum[t]; shC[t] = pCnt[t]; shM[t] = pMax[t];
    __syncthreads();
    for (int o = 256; o > 0; o >>= 1) {
        if (t < o) {
            shS[t] += shS[t + o];
            shC[t] += shC[t + o];
            shM[t] = fmaxf(shM[t], shM[t + o]);
        }
        __syncthreads();
    }
    if (t == 0) {
        float maxArg = shM[0];
        float eMax = __expf(maxArg);
        scal[0] = shS[0] + shC[0] * eMax;  // S
        scal[1] = maxArg;
    }
}

// init: V = 1/S (initial global normalization), U = 1, zero rv partials
__global__ void init_kernel(float* __restrict__ V, float* __restrict__ U,
                            float* __restrict__ rvP, const float* __restrict__ scal) {
    int i = blockIdx.x * 256 + threadIdx.x;  // [0, 16384)
    float S = scal[0];
    V[i] = 1.0f / S;
    if (i < K_DIM) U[i] = 1.0f;
#pragma unroll
    for (int j = 0; j < (SLABS * K_DIM) / B_DIM; ++j)  // 6
        rvP[i + j * B_DIM] = 0.f;
}

// ---------------------------------------------------------------------------
// R1: rv[k] = sum_b E[b,k]*V[b] via V_WMMA_F32_16X16X4_F32 ones-matrix trick.
// A = ones(16x4); B[kk,n] = E(row)*V(row) -> every D column = 4-row col-sum,
// chained as C accumulator over a 512-row slab. Invariant to the exact
// kk<->lane mapping of B because A is all-ones. Writes per-(slab,k) partials
// (no atomics -> deterministic).
// ---------------------------------------------------------------------------
__global__ void r1_colsum_wmma(const float* __restrict__ Q, const float* __restrict__ V,
                               float* __restrict__ rvP, const float* __restrict__ scal) {
    float eMax = __expf(scal[1]);
    int lane = threadIdx.x & 31;
    int wave = threadIdx.x >> 5;          // 8 waves, 16 k's each
    int n = lane & 15;
    int h = lane >> 4;                    // half-wave: rows {0,1} vs {2,3} of each chunk
    int col = blockIdx.x * 128 + wave * 16 + n;
    int slab = blockIdx.y;                // 32 slabs x 512 rows
    int b0 = slab * 512 + 2 * h;

    v2f a; a[0] = 1.0f; a[1] = 1.0f;      // ones A-matrix (16x4)
    v8f c = {};                           // accumulator

    const float* qp = Q + b0 * K_DIM + col;
    const float* vp = V + b0;
    for (int it = 0; it < 128; ++it) {    // 128 chunks of 4 rows = 512 rows
        float q0 = qp[0];
        float q1 = qp[K_DIM];
        float va = vp[0];
        float vb = vp[1];
        v2f bm;
        bm[0] = expClamped(q0, eMax) * va;
        bm[1] = expClamped(q1, eMax) * vb;
        c = __builtin_amdgcn_wmma_f32_16x16x4_f32(false, a, false, bm,
                                                  (short)0, c, false, false);
        qp += 4 * K_DIM;
        vp += 4;
    }
    // D[m,n] identical for all m; lane n (n<16), VGPR0 holds column k0+n sum
    if (lane < 16) rvP[slab * K_DIM + col] = c[0];
}

// U update: rowsum = U*rv ; u = r/rowsum ; replace inf(u) with max finite u ;
// U *= u.  Single block, fixed-order partial reduction over the 32 slabs.
__global__ void u_update(float* __restrict__ U, const float* __restrict__ rvP) {
    __shared__ float sh[1024];
    int t = threadIdx.x;
    float uu[3];
    int isI[3];
    float lmax = 0.f;
#pragma unroll
    for (int j = 0; j < 3; ++j) {
        int k = t + j * 1024;
        float rv = 0.f;
        for (int s = 0; s < SLABS; ++s) rv += rvP[s * K_DIM + k];
        float rowsum = U[k] * rv;
        float x = R_TGT / rowsum;
        int ii = (__builtin_fabsf(x) == __builtin_inff());
        uu[j] = x;
        isI[j] = ii;
        lmax = fmaxf(lmax, ii ? 0.f : x);  // max over inf-zeroed array (ref semantics)
    }
    sh[t] = lmax;
    __syncthreads();
    for (int o = 512; o > 0; o >>= 1) {
        if (t < o) sh[t] = fmaxf(sh[t], sh[t + o]);
        __syncthreads();
    }
    float m = sh[0];
#pragma unroll
    for (int j = 0; j < 3; ++j) {
        int k = t + j * 1024;
        U[k] *= (isI[j] ? m : uu[j]);
    }
}

// R2: cu[b] = sum_k E[b,k]*U[k]. Wave-per-row, b128 coalesced loads,
// wave32 shuffle reduction. Prefetch demonstrates global_prefetch_b8.
__global__ void r2_rowdot(const float* __restrict__ Q, const float* __restrict__ U,
                          float* __restrict__ cu, const float* __restrict__ scal) {
    float eMax = __expf(scal[1]);
    int wave = threadIdx.x >> 5;
    int lane = threadIdx.x & 31;
    int row = blockIdx.x * 8 + wave;
    const float4* q4 = (const float4*)(Q + (size_t)row * K_DIM);
    const float4* u4 = (const float4*)U;
    float acc = 0.f;
    for (int i = lane; i < K_DIM / 4; i += 32) {
        if (i + 32 < K_DIM / 4) __builtin_prefetch(q4 + i + 32, 0, 0);
        float4 q = q4[i];
        float4 u = u4[i];
        acc += expClamped(q.x, eMax) * u.x;
        acc += expClamped(q.y, eMax) * u.y;
        acc += expClamped(q.z, eMax) * u.z;
        acc += expClamped(q.w, eMax) * u.w;
    }
    for (int off = 16; off > 0; off >>= 1) acc += __shfl_down(acc, off, 32);
    if (lane == 0) cu[row] = acc;
}

// V update: V = c / cu (no inf-replacement in reference for column step);
// also re-zero the rv partial buffer for the next iteration.
__global__ void v_update(float* __restrict__ V, const float* __restrict__ cu,
                         float* __restrict__ rvP) {
    int i = blockIdx.x * 256 + threadIdx.x;  // [0, 16384)
    V[i] = C_TGT / cu[i];
#pragma unroll
    for (int j = 0; j < (SLABS * K_DIM) / B_DIM; ++j)
        rvP[i + j * B_DIM] = 0.f;
}

// Final: after iter-3 row scaling, colsum[b] = V[b]*cu[b]; V cancels:
// Out[b,k] = E[b,k]*U[k]/cu[b]  (also performs the transpose back to (B,K)).
__global__ void finalize_kernel(const float4* __restrict__ Q4, const float* __restrict__ U,
                                const float* __restrict__ cu, float4* __restrict__ Out4,
                                const float* __restrict__ scal) {
    float eMax = __expf(scal[1]);
    int b = blockIdx.y;
    int k4 = blockIdx.x * 256 + threadIdx.x;  // [0, 768)
    int idx = b * (K_DIM / 4) + k4;
    float4 q = Q4[idx];
    float4 u = ((const float4*)U)[k4];
    float inv = 1.0f / cu[b];
    float4 o;
    o.x = expClamped(q.x, eMax) * u.x * inv;
    o.y = expClamped(q.y, eMax) * u.y * inv;
    o.z = expClamped(q.z, eMax) * u.z * inv;
    o.w = expClamped(q.w, eMax) * u.w * inv;
    Out4[idx] = o;
}

// ---------------------------------------------------------------------------
extern "C" void kernel_launch(void* const* d_in, const int* in_sizes, int n_in,
                              void* d_out, int out_size, void* d_ws, size_t ws_size,
                              hipStream_t stream) {
    (void)in_sizes; (void)n_in; (void)out_size; (void)ws_size;
    const float* Q = (const float*)d_in[0];
    float* out = (float*)d_out;
    float* ws = (float*)d_ws;

    // workspace layout (floats), total ~545 KB
    float* scal = ws;                 // [0]=S, [1]=maxFiniteArg
    float* pSum = ws + 16;            // 512
    float* pCnt = ws + 16 + 512;      // 512
    float* pMax = ws + 16 + 1024;     // 512
    float* U    = ws + 2048;          // 3072
    float* V    = U + K_DIM;          // 16384
    float* cu   = V + B_DIM;          // 16384
    float* rvP  = cu + B_DIM;         // 32*3072 = 98304

    stats_kernel<<<512, 256, 0, stream>>>((const float4*)Q, pSum, pCnt, pMax);
    reduce0_kernel<<<1, 512, 0, stream>>>(pSum, pCnt, pMax, scal);
    init_kernel<<<64, 256, 0, stream>>>(V, U, rvP, scal);

    for (int it = 0; it < 3; ++it) {
        r1_colsum_wmma<<<dim3(K_DIM / 128, SLABS), 256, 0, stream>>>(Q, V, rvP, scal);
        u_update<<<1, 1024, 0, stream>>>(U, rvP);
        r2_rowdot<<<B_DIM / 8, 256, 0, stream>>>(Q, U, cu, scal);
        if (it < 2) v_update<<<64, 256, 0, stream>>>(V, cu, rvP);
    }

    finalize_kernel<<<dim3(3, B_DIM), 256, 0, stream>>>((const float4*)Q, U, cu,
                                                        (float4*)out, scal);
}